// E3NNLifting_21251498181472
// MI455X (gfx1250) — compile-verified
//
#include <hip/hip_runtime.h>

typedef __attribute__((ext_vector_type(2))) float v2f;
typedef __attribute__((ext_vector_type(8))) float v8f;

// out = x (N x 4) @ M (4 x 128), M built from w1/w0:
//   M[k, j*3+k] = w1[j] (k<3),  M[3, 126+m] = w0[m], else 0.
// One wave computes 16-row x 128-col output tiles with 8x V_WMMA_F32_16X16X4_F32,
// looping grid-stride over tiles so the B-tiles (weights) stay in registers.
__global__ __launch_bounds__(256) void e3nn_lift_wmma(
    const float* __restrict__ x,
    const float* __restrict__ w1,
    const float* __restrict__ w0,
    float* __restrict__ out,
    int ntiles)
{
    const int lane   = threadIdx.x & 31;
    const int wave   = threadIdx.x >> 5;
    const int gwave  = blockIdx.x * 8 + wave;        // global wave id
    const int nwaves = gridDim.x * 8;

    const int n  = lane & 15;   // col within 16-wide tile / row of A
    const int hi = lane >> 4;   // half-wave select
    const int k0 = hi * 2;      // first K held by this lane's VGPR pair

    // ---- Build the 8 B-tiles (4x16 column slices of M) once per wave. ----
    // B layout (32-bit, 4x16): VGPR0: lanes0-15 row K=0, lanes16-31 row K=2;
    //                          VGPR1: lanes0-15 row K=1, lanes16-31 row K=3.
    v2f b[8];
#pragma unroll
    for (int t = 0; t < 8; ++t) {
        const int c = t * 16 + n;
        float v0 = 0.0f, v1 = 0.0f;
        if (c < 126) {
            const float w  = w1[c / 3];
            const int   kk = c % 3;
            if (kk == k0)     v0 = w;
            if (kk == k0 + 1) v1 = w;
        } else {
            if (k0 + 1 == 3)  v1 = w0[c - 126];   // only K=3 row nonzero here
        }
        b[t].x = v0;
        b[t].y = v1;
    }

    // ---- Grid-stride loop over 16-row tiles (condition is wave-uniform). ----
    for (int tile = gwave; tile < ntiles; tile += nwaves) {
        // A layout (32-bit, 16x4): lanes0-15: VGPR0=K0, VGPR1=K1;
        //                          lanes16-31: VGPR0=K2, VGPR1=K3.  M = lane&15.
        const float* xrow = x + (size_t)tile * 64;       // 16 rows * 4 floats
        const v2f a = *(const v2f*)(xrow + n * 4 + k0);  // contiguous 256B/wave

        float* obase = out + (size_t)tile * 16 * 128;

#pragma unroll
        for (int t = 0; t < 8; ++t) {
            v8f c = {};
            // 8 args: (neg_a, A, neg_b, B, c_mod, C, reuse_a, reuse_b)
            c = __builtin_amdgcn_wmma_f32_16x16x4_f32(
                    false, a, false, b[t], (short)0, c, false, false);

            // D layout: VGPR r, lane L -> row = r + 8*hi, col = n (tile t).
            // Output is streamed (512 MB >> L2): non-temporal stores.
#pragma unroll
            for (int r = 0; r < 8; ++r) {
                __builtin_nontemporal_store(
                    c[r], obase + (size_t)(r + 8 * hi) * 128 + t * 16 + n);
            }
        }
    }
}

// Scalar tail for row counts not divisible by 16 (not hit for N = 1e6).
__global__ void e3nn_lift_tail(
    const float* __restrict__ x,
    const float* __restrict__ w1,
    const float* __restrict__ w0,
    float* __restrict__ out,
    int row0, int nrows)
{
    const int idx  = blockIdx.x * blockDim.x + threadIdx.x;
    const int rows = nrows - row0;
    if (idx >= rows * 128) return;
    const int row = row0 + idx / 128;
    const int c   = idx % 128;
    float v;
    if (c < 126) v = x[(size_t)row * 4 + (c % 3)] * w1[c / 3];
    else         v = x[(size_t)row * 4 + 3] * w0[c - 126];
    out[(size_t)row * 128 + c] = v;
}

extern "C" void kernel_launch(void* const* d_in, const int* in_sizes, int n_in,
                              void* d_out, int out_size, void* d_ws, size_t ws_size,
                              hipStream_t stream) {
    const float* x  = (const float*)d_in[0];
    const float* w1 = (const float*)d_in[1];
    const float* w0 = (const float*)d_in[2];
    float* out = (float*)d_out;

    const int nrows  = in_sizes[0] / 4;
    const int ntiles = nrows / 16;

    if (ntiles > 0) {
        int blocks = 2048;                        // 16384 waves, grid-stride
        const int maxBlocks = (ntiles + 7) / 8;
        if (blocks > maxBlocks) blocks = maxBlocks;
        e3nn_lift_wmma<<<blocks, 256, 0, stream>>>(x, w1, w0, out, ntiles);
    }
    const int rem = nrows - ntiles * 16;
    if (rem > 0) {
        const int total = rem * 128;
        e3nn_lift_tail<<<(total + 255) / 256, 256, 0, stream>>>(
            x, w1, w0, out, ntiles * 16, nrows);
    }
}